// GptOssTopKRouter_19980187861075
// MI455X (gfx1250) — compile-verified
//
#include <hip/hip_runtime.h>
#include <hip/hip_bf16.h>
#include <math.h>

// Problem constants (match reference)
#define T_TOKENS 16384   // BATCH * SEQ = 4 * 4096
#define HIDDEN   2880
#define NEXP     128
#define TOPK     4

// Tiling
#define MT   256                  // tokens per workgroup (2 M-tiles per wave)
#define KC   32                   // K-chunk staged in LDS
#define ROWF 36                   // padded row stride in floats (TDM pad: 32 dw + 4 dw)
#define NCH  (HIDDEN / KC)        // 90 chunks
#define BUFF ((MT + NEXP) * ROWF) // floats per double-buffer slot (13824)
#define CPAD 132                  // padded row stride for C write-out tile
// dynamic LDS: max(double buffers, C tile)
#define SMEM_FLOATS ((2 * BUFF) > (MT * CPAD) ? (2 * BUFF) : (MT * CPAD))

typedef __attribute__((ext_vector_type(2))) float v2f;
typedef __attribute__((ext_vector_type(8))) float v8f;
typedef unsigned int u32;
typedef __attribute__((ext_vector_type(4))) u32 v4u;
typedef __attribute__((ext_vector_type(8))) int v8i;
typedef __attribute__((ext_vector_type(4))) int v4i;

// ---------------------------------------------------------------------------
// TDM: 2-D tile load Global -> LDS with hardware row padding.
// Tile: tile_d0 (contiguous f32) x tile_d1 rows, row stride = stride0 elems.
// pad: insert 4 DWORDs after every 32 DWORDs -> LDS row stride = 36 floats.
// ---------------------------------------------------------------------------
__device__ static inline void tdm_load_2d_f32(u32 lds_off, const float* gaddr,
                                              u32 tile_d0, u32 tile_d1,
                                              u32 tensor_d0, u32 tensor_d1,
                                              u32 stride0) {
  const unsigned long long ga = (unsigned long long)(uintptr_t)gaddr;
  v4u g0;
  g0.x = 1u;                                               // count=1 (valid), user mode
  g0.y = lds_off;                                          // lds_addr (bytes)
  g0.z = (u32)ga;                                          // global_addr[31:0]
  g0.w = (u32)((ga >> 32) & 0x01FFFFFFu) | (2u << 30);     // global_addr[56:32] | type=2
  v8i g1;
  g1[0] = (int)((2u << 16)      // data_size = 4 bytes
              | (1u << 20)      // pad_enable
              | (4u << 22)      // pad_interval: 32 DWORDs
              | (3u << 25));    // pad_amount: 4 DWORDs
  g1[1] = (int)((tensor_d0 & 0xFFFFu) << 16);                              // dim0[15:0]
  g1[2] = (int)(((tensor_d0 >> 16) & 0xFFFFu) | ((tensor_d1 & 0xFFFFu) << 16));
  g1[3] = (int)(((tensor_d1 >> 16) & 0xFFFFu) | ((tile_d0 & 0xFFFFu) << 16));
  g1[4] = (int)(tile_d1 & 0xFFFFu);                        // tile_dim1 | tile_dim2=0
  g1[5] = (int)stride0;                                    // tensor_dim0_stride[31:0]
  g1[6] = 0;                                               // stride hi | dim1_stride lo
  g1[7] = 0;
  const v4i z4 = {0, 0, 0, 0};
  const v8i z8 = {0, 0, 0, 0, 0, 0, 0, 0};
  __builtin_amdgcn_tensor_load_to_lds(g0, g1, z4, z4, z8, 0);
}

// ---------------------------------------------------------------------------
// Kernel 1: router logits GEMM via V_WMMA_F32_16X16X4_F32 with TDM
// double-buffered staging. 8 wave32 waves; each wave: 32 token rows x 128
// experts (2 M-tiles x 8 N-tiles) so every B fragment feeds 2 WMMAs.
// ---------------------------------------------------------------------------
__global__ __launch_bounds__(256) void router_gemm_wmma(
    const float* __restrict__ X,      // [T, H]
    const float* __restrict__ W,      // [E, H]
    const float* __restrict__ Bias,   // [E]
    float* __restrict__ logits)       // [T, E]
{
  extern __shared__ float smem[];

  const int tid    = threadIdx.x;
  const int wave   = tid >> 5;
  const int lane   = tid & 31;
  const int laneLo = lane & 15;
  const int kOff   = (lane >> 4) << 1;   // lanes 16-31 hold K+2,K+3 (ISA 16x4 f32 layout)
  const long rowBase = (long)blockIdx.x * MT;

  const u32 ldsBase = (u32)(uintptr_t)(void*)smem;   // LDS aperture low bits = offset

  v8f acc0[8] = {};                      // rows wave*32 +  0..15, 8 N-tiles
  v8f acc1[8] = {};                      // rows wave*32 + 16..31, 8 N-tiles

  // Preload chunk 0 into buffer 0 (single wave issues the DMA).
  if (wave == 0) {
    tdm_load_2d_f32(ldsBase, X + rowBase * (long)HIDDEN,
                    KC, MT, HIDDEN, MT, HIDDEN);
    tdm_load_2d_f32(ldsBase + (u32)(MT * ROWF * 4), W,
                    KC, NEXP, HIDDEN, NEXP, HIDDEN);
  }

  for (int c = 0; c < NCH; ++c) {
    if (wave == 0) __builtin_amdgcn_s_wait_tensorcnt(0);   // chunk c landed
    __syncthreads();                                        // publish buf[c&1]

    // Kick off chunk c+1 into the other buffer while we compute on this one.
    if (wave == 0 && (c + 1) < NCH) {
      const u32 nb = ldsBase + (u32)(((c + 1) & 1) * BUFF * 4);
      const int k0 = (c + 1) * KC;
      tdm_load_2d_f32(nb, X + rowBase * (long)HIDDEN + k0,
                      KC, MT, HIDDEN, MT, HIDDEN);
      tdm_load_2d_f32(nb + (u32)(MT * ROWF * 4), W + k0,
                      KC, NEXP, HIDDEN, NEXP, HIDDEN);
    }

    const float* bX = smem + (c & 1) * BUFF;
    const float* bW = bX + MT * ROWF;

    #pragma unroll
    for (int kk = 0; kk < KC; kk += 4) {
      const int col = kk + kOff;
      const v2f a0 = *(const v2f*)&bX[(wave * 32 + laneLo) * ROWF + col];
      const v2f a1 = *(const v2f*)&bX[(wave * 32 + 16 + laneLo) * ROWF + col];
      v2f b[8];
      #pragma unroll
      for (int t = 0; t < 8; ++t)
        b[t] = *(const v2f*)&bW[(t * 16 + laneLo) * ROWF + col];
      #pragma unroll
      for (int t = 0; t < 8; ++t) {
        acc0[t] = __builtin_amdgcn_wmma_f32_16x16x4_f32(
            false, a0, false, b[t], (short)0, acc0[t], false, false);
        acc1[t] = __builtin_amdgcn_wmma_f32_16x16x4_f32(
            false, a1, false, b[t], (short)0, acc1[t], false, false);
      }
    }
    // No trailing barrier needed: next iteration's top barrier orders
    // compute(c) before TDM writes into buf[c&1] at iteration c+1 (chunk c+2).
  }

  // ---- Epilogue: spill C (+bias) to LDS, then coalesced float4 stores ----
  __syncthreads();                       // all chunk compute done; smem reusable
  float* sC = smem;                      // [MT][CPAD] tile
  const int mBase = (lane >> 4) << 3;    // 0 or 8 (C layout rows)
  #pragma unroll
  for (int t = 0; t < 8; ++t) {
    const int n = t * 16 + laneLo;
    const float bb = Bias[n];
    #pragma unroll
    for (int e = 0; e < 8; ++e) {
      sC[(wave * 32 +      mBase + e) * CPAD + n] = acc0[t][e] + bb;
      sC[(wave * 32 + 16 + mBase + e) * CPAD + n] = acc1[t][e] + bb;
    }
  }
  __syncthreads();

  // MT rows x 128 floats = MT*32 float4; 32 per thread, contiguous per row.
  #pragma unroll
  for (int i = 0; i < (MT * 32) / 256; ++i) {
    const int idx = tid + 256 * i;       // 0..MT*32-1
    const int r   = idx >> 5;            // row
    const int c4  = (idx & 31) << 2;     // float offset in row
    const float4 val = *(const float4*)&sC[r * CPAD + c4];
    *(float4*)&logits[(rowBase + r) * NEXP + c4] = val;
  }
}

// ---------------------------------------------------------------------------
// Kernel 2: per-token top-4 + softmax. One wave32 per token.
// ---------------------------------------------------------------------------
__global__ __launch_bounds__(256) void topk_softmax_kernel(
    const float* __restrict__ logits,   // [T, E]
    float* __restrict__ rw,             // [T, TOPK] routing weights
    int*   __restrict__ se)             // [T, TOPK] selected experts
{
  const int lane  = threadIdx.x & 31;
  const int wave  = threadIdx.x >> 5;
  const int token = blockIdx.x * 8 + wave;

  float v[4];
  #pragma unroll
  for (int j = 0; j < 4; ++j)
    v[j] = logits[(long)token * NEXP + lane + 32 * j];

  float topv[TOPK];
  int   topi[TOPK];

  #pragma unroll
  for (int s = 0; s < TOPK; ++s) {
    float bv = -INFINITY;
    int   bi = 0x7fffffff;
    #pragma unroll
    for (int j = 0; j < 4; ++j) {
      const int idx = lane + 32 * j;
      if (v[j] > bv || (v[j] == bv && idx < bi)) { bv = v[j]; bi = idx; }
    }
    #pragma unroll
    for (int off = 16; off >= 1; off >>= 1) {
      const float ov = __shfl_xor(bv, off, 32);
      const int   oi = __shfl_xor(bi, off, 32);
      if (ov > bv || (ov == bv && oi < bi)) { bv = ov; bi = oi; }
    }
    topv[s] = bv;
    topi[s] = bi;
    #pragma unroll
    for (int j = 0; j < 4; ++j)
      if (lane + 32 * j == bi) v[j] = -INFINITY;
  }

  if (lane == 0) {
    const float m = topv[0];
    float ex[TOPK], sum = 0.f;
    #pragma unroll
    for (int s = 0; s < TOPK; ++s) { ex[s] = __expf(topv[s] - m); sum += ex[s]; }
    const float inv = 1.f / sum;
    #pragma unroll
    for (int s = 0; s < TOPK; ++s) {
      rw[(long)token * TOPK + s] = ex[s] * inv;
      se[(long)token * TOPK + s] = topi[s];
    }
  }
}

// ---------------------------------------------------------------------------
extern "C" void kernel_launch(void* const* d_in, const int* in_sizes, int n_in,
                              void* d_out, int out_size, void* d_ws, size_t ws_size,
                              hipStream_t stream) {
  const float* X    = (const float*)d_in[0];  // hidden_states [4,4096,2880]
  const float* W    = (const float*)d_in[1];  // weight        [128,2880]
  const float* Bias = (const float*)d_in[2];  // bias          [128]

  float* out    = (float*)d_out;
  float* rw     = out;                                        // [T, 4] f32
  int*   se     = (int*)(out + (size_t)T_TOKENS * TOPK);      // [T, 4] i32 bits
  float* logits = out + (size_t)2 * T_TOKENS * TOPK;          // [T, 128] f32

  const size_t smemBytes = (size_t)SMEM_FLOATS * sizeof(float);  // 135168 B
  router_gemm_wmma<<<T_TOKENS / MT, 256, smemBytes, stream>>>(X, W, Bias, logits);
  topk_softmax_kernel<<<T_TOKENS / 8, 256, 0, stream>>>(logits, rw, se);
}